// CatmullRomSpline4D_80470507258269
// MI455X (gfx1250) — compile-verified
//
#include <hip/hip_runtime.h>

// Problem constants (from reference): knots[D=16][Z=64][Y=128][X=128][C=2] f32
#define DD 16
#define ZDIM 64
#define YDIM 128
#define XDIM 128

typedef __attribute__((ext_vector_type(2))) float v2f;
typedef __attribute__((ext_vector_type(8))) float v8f;

// UNIFORM_BASIS = HERMITE @ CR:
//   row k multiplies s^(3-k);  w_j = s^3*UB[0][j] + s^2*UB[1][j] + s*UB[2][j] + UB[3][j]
//   col0: (-0.5, 1.0, -0.5, 0.0)
//   col1: ( 1.5,-2.5,  0.0, 1.0)
//   col2: (-1.5, 2.0,  0.5, 0.0)
//   col3: ( 0.5,-0.5,  0.0, 0.0)

__device__ __forceinline__ void spline_idx(float coord, int size, int id[4], float* s_out) {
    int i0 = (int)coord;                       // trunc, matches astype(int32)
    float s = coord - (float)i0;
    s = fminf(fmaxf(s, 0.0f), 1.0f);
    #pragma unroll
    for (int k = 0; k < 4; ++k) {
        int v = i0 - 1 + k;
        v = v < 0 ? 0 : (v > size - 1 ? size - 1 : v);
        id[k] = v;
    }
    *s_out = s;
}

#if defined(__gfx1250__) && __has_builtin(__builtin_amdgcn_wmma_f32_16x16x4_f32)
#define USE_WMMA_WEIGHTS 1
#endif

#ifdef USE_WMMA_WEIGHTS
// One V_WMMA_F32_16X16X4_F32 evaluates all 4 Catmull-Rom weights for the 16
// points of this wave (points = B columns). A holds the basis in rows 0-3 AND
// rows 8-11 so that lanes 0-15 (D rows 0-7) and lanes 16-31 (D rows 8-15) both
// receive w0..w3 in result elements 0..3 — no shuffles needed.
__device__ __forceinline__ v8f cr_weights_wmma(v2f A, float s, bool lo) {
    v2f B;
    // B layout (4x16): VGPR0 = K0 (lanes 0-15) / K2 (lanes 16-31); VGPR1 = K1 / K3
    B.x = lo ? s * s * s : s;
    B.y = lo ? s * s     : 1.0f;
    v8f c = {0.f, 0.f, 0.f, 0.f, 0.f, 0.f, 0.f, 0.f};
    return __builtin_amdgcn_wmma_f32_16x16x4_f32(false, A, false, B, (short)0, c,
                                                 false, false);
}
#endif

__device__ __forceinline__ void cr_weights_valu(float s, float w[4]) {
    float s2 = s * s, s3 = s2 * s;
    w[0] = -0.5f * s3 + 1.0f * s2 - 0.5f * s;
    w[1] =  1.5f * s3 - 2.5f * s2 + 1.0f;
    w[2] = -1.5f * s3 + 2.0f * s2 + 0.5f * s;
    w[3] =  0.5f * s3 - 0.5f * s2;
}

__global__ __launch_bounds__(256)
void spline4d_kernel(const float* __restrict__ idx,
                     const float* __restrict__ knots,
                     const float* __restrict__ depth_p,
                     float* __restrict__ out, int n) {
    const int lane = threadIdx.x & 31;
    const int wave = threadIdx.x >> 5;
    const bool lo  = lane < 16;                 // low half handles depth slabs 0,1; high half 2,3
    int p = blockIdx.x * 128 + wave * 16 + (lane & 15);
    const bool valid = p < n;
    if (p >= n) p = n - 1;                      // keep EXEC all-ones for WMMA; mask the store

    // --- coordinates ---
    float cz_c = idx[3 * p + 0];
    float cy_c = idx[3 * p + 1];
    float cx_c = idx[3 * p + 2];
    float depth = depth_p[0];
    int ind = (int)floorf(depth);
    ind = ind < 1 ? 1 : (ind > DD - 1 ? DD - 1 : ind);
    float dc = (float)(ind - 1) + (depth - (float)ind);  // == depth-1 in-range

    int idp[4], izv[4], iyv[4], ixv[4];
    float sd, sz, sy, sx;
    spline_idx(dc,   DD,   idp, &sd);
    spline_idx(cz_c, ZDIM, izv, &sz);
    spline_idx(cy_c, YDIM, iyv, &sy);
    spline_idx(cx_c, XDIM, ixv, &sx);

    float wd[4], wz[4], wy[4], wx[4];
#ifdef USE_WMMA_WEIGHTS
    {
        // Build A (16x4): A[m][k] = UB[k][m%4] for m in {0..3, 8..11}, else 0.
        // A VGPR layout: lane = row M (lane&15); VGPR0 = K0/K2, VGPR1 = K1/K3.
        int row = lane & 15, j = row & 3;
        bool active = (row < 4) || (row >= 8 && row < 12);
        float c0 = j == 0 ? -0.5f : j == 1 ?  1.5f : j == 2 ? -1.5f :  0.5f;
        float c1 = j == 0 ?  1.0f : j == 1 ? -2.5f : j == 2 ?  2.0f : -0.5f;
        float c2 = j == 0 ? -0.5f : j == 1 ?  0.0f : j == 2 ?  0.5f :  0.0f;
        float c3 = j == 0 ?  0.0f : j == 1 ?  1.0f : j == 2 ?  0.0f :  0.0f;
        v2f A;
        A.x = active ? (lo ? c0 : c2) : 0.0f;
        A.y = active ? (lo ? c1 : c3) : 0.0f;

        v8f rd = cr_weights_wmma(A, sd, lo);
        v8f rz = cr_weights_wmma(A, sz, lo);
        v8f ry = cr_weights_wmma(A, sy, lo);
        v8f rx = cr_weights_wmma(A, sx, lo);
        #pragma unroll
        for (int k = 0; k < 4; ++k) { wd[k] = rd[k]; wz[k] = rz[k]; wy[k] = ry[k]; wx[k] = rx[k]; }
    }
#else
    cr_weights_valu(sd, wd);
    cr_weights_valu(sz, wz);
    cr_weights_valu(sy, wy);
    cr_weights_valu(sx, wx);
#endif

    // --- gather + contract: this lane handles 2 of the 4 depth slabs ---
    const float2* __restrict__ kn = (const float2*)knots;   // [D][Z][Y][X] of float2 (C=2)
    int   dIdx0 = lo ? idp[0] : idp[2];
    int   dIdx1 = lo ? idp[1] : idp[3];
    float dW0   = lo ? wd[0]  : wd[2];
    float dW1   = lo ? wd[1]  : wd[3];
    int   dIdx[2] = {dIdx0, dIdx1};
    float dW[2]   = {dW0, dW1};

    float acc0 = 0.0f, acc1 = 0.0f;
    #pragma unroll
    for (int dd = 0; dd < 2; ++dd) {
        const float w_d = dW[dd];
        const int ofd = dIdx[dd] * ZDIM;
        #pragma unroll
        for (int z = 0; z < 4; ++z) {
            const float wdz = w_d * wz[z];
            const int ofz = (ofd + izv[z]) * YDIM;
            #pragma unroll
            for (int y = 0; y < 4; ++y) {
                const float wdzy = wdz * wy[y];
                const float2* rowp = kn + (size_t)(ofz + iyv[y]) * XDIM;
                float2 k0 = rowp[ixv[0]];
                float2 k1 = rowp[ixv[1]];
                float2 k2 = rowp[ixv[2]];
                float2 k3 = rowp[ixv[3]];
                float s0 = wx[0] * k0.x + wx[1] * k1.x + wx[2] * k2.x + wx[3] * k3.x;
                float s1 = wx[0] * k0.y + wx[1] * k1.y + wx[2] * k2.y + wx[3] * k3.y;
                acc0 = fmaf(wdzy, s0, acc0);
                acc1 = fmaf(wdzy, s1, acc1);
            }
        }
    }

    // combine the two half-slab lanes (wave32: xor 16 pairs lane n with n+16)
    acc0 += __shfl_xor(acc0, 16);
    acc1 += __shfl_xor(acc1, 16);

    if (lo && valid) {
        ((float2*)out)[p] = make_float2(acc0, acc1);
    }
}

extern "C" void kernel_launch(void* const* d_in, const int* in_sizes, int n_in,
                              void* d_out, int out_size, void* d_ws, size_t ws_size,
                              hipStream_t stream) {
    (void)n_in; (void)out_size; (void)d_ws; (void)ws_size;
    const float* idx   = (const float*)d_in[0];   // [N,3] f32
    const float* knots = (const float*)d_in[1];   // [16,64,128,128,2] f32
    const float* depth = (const float*)d_in[2];   // scalar f32
    float* out = (float*)d_out;                   // [N,2] f32
    int n = in_sizes[0] / 3;
    int blocks = (n + 127) / 128;                 // 128 points per 256-thread block
    spline4d_kernel<<<blocks, 256, 0, stream>>>(idx, knots, depth, out, n);
}